// NodeNet_77137612636506
// MI455X (gfx1250) — compile-verified
//
#include <hip/hip_runtime.h>

// MI455X / gfx1250, wave32. Bandwidth-bound problem (2x512MB matrices, each
// read twice = 2GB @ 23.3 TB/s ~= 86us floor). Math collapsed to:
//   out = Ri @ (e .* (Ro^T @ (X k0))) + Ro @ (e .* (Ri^T @ (X k1))) + X k2
// Dot-product accumulation via V_WMMA_F32_16X16X4_F32: one operand is the
// streamed R tile (256B per WMMA), the other a replicated 4-vector slice.
// R-stream loads use TH=NT (non-temporal) so the hot u/s broadcast tables
// stay resident in WGP$/L2 while 1 GB/pass streams through.

typedef __attribute__((ext_vector_type(2))) float v2f;
typedef __attribute__((ext_vector_type(8))) float v8f;

// ---------------- K0: u0 = X k0, u1 = X k1, ub = X k2 -----------------------
__global__ __launch_bounds__(256) void k0_prepare(const float* __restrict__ X,
    const float* __restrict__ kern, float* __restrict__ u0,
    float* __restrict__ u1, float* __restrict__ ub, int N) {
  int n = blockIdx.x * blockDim.x + threadIdx.x;
  if (n >= N) return;
  float4 x = *(const float4*)(X + 4 * n);
  u0[n] = x.x * kern[0] + x.y * kern[1] + x.z * kern[2]  + x.w * kern[3];
  u1[n] = x.x * kern[4] + x.y * kern[5] + x.z * kern[6]  + x.w * kern[7];
  ub[n] = x.x * kern[8] + x.y * kern[9] + x.z * kern[10] + x.w * kern[11];
}

// ---------------- K1: tpart[mat][nc][j] = partial (R^T u)[j] ----------------
// mat 0: R=Ro with u0 ; mat 1: R=Ri with u1.
// Per WMMA: A = u[n0..n0+3] replicated over 16 rows (doc layout: lane<16 ->
// K0,K1 ; lane>=16 -> K2,K3), B = 4 rows x 16 cols of R (mirror layout:
// lane l holds column j0+(l&15), VGPR0=K(0|2), VGPR1=K(1|3)). C row 0 gives
// 16 partial column sums.
__global__ __launch_bounds__(256) void k1_colsum(const float* __restrict__ Ri,
    const float* __restrict__ Ro, const float* __restrict__ u0,
    const float* __restrict__ u1, float* __restrict__ tpart, int N, int E) {
  const int lane = threadIdx.x & 31;
  const int wid  = blockIdx.x * (blockDim.x >> 5) + (threadIdx.x >> 5);
  const int JT   = E >> 4;                 // #16-wide edge tiles
  const int jt   = wid % JT;               // consecutive waves -> adjacent tiles
  const int nc   = (wid / JT) & 7;         // 8 row chunks
  const int mat  = wid / (JT * 8);
  const float* __restrict__ R = mat ? Ri : Ro;
  const float* __restrict__ u = mat ? u1 : u0;
  const int j0   = jt << 4;
  const int rows = N >> 3;                 // rows per chunk (1024)
  const int nb   = nc * rows;
  const int half = lane >> 4;              // 0: K=0,1  1: K=2,3
  const int lcol = lane & 15;
  const float* rp = R + (size_t)(nb + 2 * half) * E + j0 + lcol;
  const float* up = u + nb + 2 * half;
  v8f acc = {};
  const int iters = rows >> 2;             // 4 K-values per WMMA
  for (int it = 0; it < iters; ++it) {
    v2f a = *(const v2f*)up;               // u[n0+2h], u[n0+2h+1]  (hot, cached)
    v2f b;
    b.x = __builtin_nontemporal_load(rp);      // R[n0+2h  , j0+lcol]  (stream)
    b.y = __builtin_nontemporal_load(rp + E);  // R[n0+2h+1, j0+lcol]  (stream)
    acc = __builtin_amdgcn_wmma_f32_16x16x4_f32(false, a, false, b,
                                                (short)0, acc, false, false);
    up += 4;
    rp += (size_t)4 * E;
  }
  // C VGPR0, lanes 0-15 = (M=0, col=lane) -> t_partial[j0+lane]
  if (lane < 16)
    tpart[(size_t)(mat * 8 + nc) * E + j0 + lane] = acc[0];
}

// ---------------- K2: so = e .* sum(tpart[0]), si = e .* sum(tpart[1]) ------
__global__ __launch_bounds__(256) void k2_scale(const float* __restrict__ e,
    const float* __restrict__ tpart, float* __restrict__ so,
    float* __restrict__ si, int E) {
  int j = blockIdx.x * blockDim.x + threadIdx.x;
  if (j >= E) return;
  float t0 = 0.f, t1 = 0.f;
#pragma unroll
  for (int nc = 0; nc < 8; ++nc) {
    t0 += __builtin_nontemporal_load(&tpart[(size_t)nc * E + j]);
    t1 += __builtin_nontemporal_load(&tpart[(size_t)(8 + nc) * E + j]);
  }
  float ev = e[j];
  so[j] = ev * t0;
  si[j] = ev * t1;
}

// ---------------- K3: opart[mat][jc][n] = partial (R s)[n] ------------------
// mat 0: R=Ri with so ; mat 1: R=Ro with si.
// Per WMMA: A = 16 rows x 4 cols of R (doc layout: lanes l and l+16 hold row
// n0+(l&15); b64 loads stream each row sequentially), B = s[j0..j0+3]
// replicated over 16 cols. C column 0 gives 16 partial row dots.
__global__ __launch_bounds__(256) void k3_rowdot(const float* __restrict__ Ri,
    const float* __restrict__ Ro, const float* __restrict__ so,
    const float* __restrict__ si, float* __restrict__ opart, int N, int E) {
  const int lane = threadIdx.x & 31;
  const int wid  = blockIdx.x * (blockDim.x >> 5) + (threadIdx.x >> 5);
  const int NT   = N >> 4;                 // #16-row node tiles
  const int nt   = wid % NT;
  const int jc   = (wid / NT) & 15;        // 16 edge chunks
  const int mat  = wid / (NT * 16);
  const float* __restrict__ R = mat ? Ro : Ri;
  const float* __restrict__ s = mat ? si : so;
  const int n0   = nt << 4;
  const int jlen = E >> 4;                 // edges per chunk (1024)
  const int jb   = jc * jlen;
  const int half = lane >> 4;
  const int lrow = lane & 15;
  const float* ap = R + (size_t)(n0 + lrow) * E + jb + 2 * half;
  const float* sp = s + jb + 2 * half;
  v8f acc = {};
  const int iters = jlen >> 2;
  for (int it = 0; it < iters; ++it) {
    v2f a = __builtin_nontemporal_load((const v2f*)ap); // R stream (b64/lane)
    v2f b = *(const v2f*)sp;               // s[j0+2h], s[j0+2h+1] (hot, cached)
    acc = __builtin_amdgcn_wmma_f32_16x16x4_f32(false, a, false, b,
                                                (short)0, acc, false, false);
    ap += 4;
    sp += 4;
  }
  // C VGPR v: lanes 0-15 -> M=v, lanes 16-31 -> M=8+v (all cols identical).
  // lanes 0-7 emit m=lane, lanes 16-23 emit m=8+(lane-16).
  int idx = (lane < 8) ? lane : ((lane >= 16 && lane < 24) ? (lane - 16) : -1);
  if (idx >= 0) {
    float r = acc[0];
#pragma unroll
    for (int vv = 1; vv < 8; ++vv) r = (idx == vv) ? acc[vv] : r;
    int m = (lane < 8) ? lane : (8 + lane - 16);
    opart[(size_t)(mat * 16 + jc) * N + n0 + m] = r;
  }
}

// ---------------- K4: out[n] = ub[n] + sum of 32 opart slabs ----------------
__global__ __launch_bounds__(256) void k4_final(const float* __restrict__ ub,
    const float* __restrict__ opart, float* __restrict__ out, int N) {
  int n = blockIdx.x * blockDim.x + threadIdx.x;
  if (n >= N) return;
  float sum = ub[n];
#pragma unroll
  for (int p = 0; p < 32; ++p)
    sum += __builtin_nontemporal_load(&opart[(size_t)p * N + n]);
  out[n] = sum;
}

extern "C" void kernel_launch(void* const* d_in, const int* in_sizes, int n_in,
                              void* d_out, int out_size, void* d_ws, size_t ws_size,
                              hipStream_t stream) {
  const float* X    = (const float*)d_in[0];
  const float* e    = (const float*)d_in[1];
  const float* Ri   = (const float*)d_in[2];
  const float* Ro   = (const float*)d_in[3];
  const float* kern = (const float*)d_in[4];
  float* out = (float*)d_out;
  const int N = in_sizes[0] / 4;   // 8192
  const int E = in_sizes[1];       // 16384

  float* ws    = (float*)d_ws;
  float* u0    = ws;               // N
  float* u1    = u0 + N;           // N
  float* ub    = u1 + N;           // N
  float* so    = ub + N;           // E
  float* si    = so + E;           // E
  float* tpart = si + E;           // 2 * 8 * E
  float* opart = tpart + 16 * E;   // 2 * 16 * N   (total ~2.3 MB)

  k0_prepare<<<(N + 255) / 256, 256, 0, stream>>>(X, kern, u0, u1, ub, N);

  const int waves1 = (E / 16) * 8 * 2;   // 16384 waves
  k1_colsum<<<waves1 / 8, 256, 0, stream>>>(Ri, Ro, u0, u1, tpart, N, E);

  k2_scale<<<(E + 255) / 256, 256, 0, stream>>>(e, tpart, so, si, E);

  const int waves3 = (N / 16) * 16 * 2;  // 16384 waves
  k3_rowdot<<<waves3 / 8, 256, 0, stream>>>(Ri, Ro, so, si, opart, N, E);

  k4_final<<<(N + 255) / 256, 256, 0, stream>>>(ub, opart, out, N);
}